// GraphUnet_45286135169790
// MI455X (gfx1250) — compile-verified
//
#include <hip/hip_runtime.h>
#include <hip/hip_bf16.h>

// ---------------------------------------------------------------------------
// Graph U-Net forward for MI455X (gfx1250).
// All dense matmuls (augment Md@Md, X@W, Ahat@Y, decode) run through a single
// f32 WMMA GEMM kernel built on V_WMMA_F32_16X16X4_F32 (wave32).
// B tile is stored transposed in LDS so both A and B fragments are single
// contiguous 64-bit LDS loads per lane (no repack movs before the WMMA).
// ---------------------------------------------------------------------------

typedef __attribute__((ext_vector_type(2))) float v2f;
typedef __attribute__((ext_vector_type(8))) float v8f;

#define GEMM_BM 128
#define GEMM_BN 64
#define GEMM_KC 32

// C[M,N] = A[M,K] @ B[K,N]  (+bias per col, optional relu, optional diag zero)
__launch_bounds__(256)
__global__ void gemm_f32_wmma(const float* __restrict__ A,
                              const float* __restrict__ B,
                              float* __restrict__ C,
                              int M, int N, int K,
                              const float* __restrict__ bias,
                              int relu, int zero_diag)
{
    __shared__ float As[GEMM_BM][GEMM_KC + 1];   // row-major, +1 pad
    __shared__ float Bs[GEMM_BN][GEMM_KC + 1];   // TRANSPOSED: Bs[col][k], +1 pad

    const int tid  = threadIdx.x;
    const int lane = tid & 31;          // wave32
    const int wave = tid >> 5;          // 8 waves
    const int half = lane >> 4;         // 0: K0/K1, 1: K2/K3 (A/B frag layout)
    const int lid  = lane & 15;

    const int bm0 = blockIdx.x * GEMM_BM;
    const int bn0 = blockIdx.y * GEMM_BN;
    const int wm0 = (wave >> 1) * 32;   // 4 wave-rows x 32
    const int wn0 = (wave & 1) * 32;    // 2 wave-cols x 32

    v8f acc[2][2] = {};

    for (int k0 = 0; k0 < K; k0 += GEMM_KC) {
        // Stage A tile (BM x KC) into LDS, zero-fill out-of-bounds.
        for (int t = tid; t < GEMM_BM * GEMM_KC; t += 256) {
            int r = t >> 5, c = t & 31;
            int gr = bm0 + r, gc = k0 + c;
            As[r][c] = (gr < M && gc < K) ? A[(size_t)gr * K + gc] : 0.0f;
        }
        // Stage B tile (KC x BN) transposed into LDS: Bs[col][k].
        for (int t = tid; t < GEMM_KC * GEMM_BN; t += 256) {
            int r = t >> 6, c = t & 63;           // r = k index, c = col index
            int gr = k0 + r, gc = bn0 + c;
            Bs[c][r] = (gr < K && gc < N) ? B[(size_t)gr * N + gc] : 0.0f;
        }
        __syncthreads();

#pragma unroll
        for (int kk = 0; kk < GEMM_KC; kk += 4) {
            v2f a[2], b[2];
#pragma unroll
            for (int i = 0; i < 2; ++i) {
                int row = wm0 + i * 16 + lid;     // A 16x4: M across lanes 0-15
                a[i].x = As[row][kk + 2 * half];
                a[i].y = As[row][kk + 2 * half + 1];
                int col = wn0 + i * 16 + lid;     // B 4x16: N across lanes 0-15
                b[i].x = Bs[col][kk + 2 * half];
                b[i].y = Bs[col][kk + 2 * half + 1];
            }
#pragma unroll
            for (int i = 0; i < 2; ++i)
#pragma unroll
                for (int j = 0; j < 2; ++j)
                    acc[i][j] = __builtin_amdgcn_wmma_f32_16x16x4_f32(
                        /*neg_a=*/false, a[i], /*neg_b=*/false, b[j],
                        /*c_mod=*/(short)0, acc[i][j],
                        /*reuse_a=*/false, /*reuse_b=*/false);
        }
        __syncthreads();
    }

    // Epilogue: C layout = VGPR v -> row v + 8*half, col = lane&15.
#pragma unroll
    for (int i = 0; i < 2; ++i) {
#pragma unroll
        for (int j = 0; j < 2; ++j) {
            int col = bn0 + wn0 + j * 16 + lid;
#pragma unroll
            for (int v = 0; v < 8; ++v) {
                int row = bm0 + wm0 + i * 16 + v + 8 * half;
                if (row < M && col < N) {
                    float val = acc[i][j][v];
                    if (bias) val += bias[col];
                    if (relu) val = val > 0.0f ? val : 0.0f;
                    if (zero_diag && row == col) val = 0.0f;
                    C[(size_t)row * N + col] = val;
                }
            }
        }
    }
}

// ---------------------------------------------------------------------------
// Elementwise / reduction helper kernels
// ---------------------------------------------------------------------------

__global__ void fill_zero_kernel(float* p, size_t count) {
    size_t idx = (size_t)blockIdx.x * 256 + threadIdx.x;
    if (idx < count) p[idx] = 0.0f;
}

__global__ void scatter_edges_kernel(const int* __restrict__ ei, int E, int n,
                                     float* __restrict__ M) {
    int e = blockIdx.x * 256 + threadIdx.x;
    if (e < E) {
        int src = ei[e];
        int dst = ei[E + e];
        atomicAdd(&M[(size_t)dst * n + src], 1.0f);
    }
}

// deg[i] = sum_j A[i,j] with improved self-loop fix; d[i] = deg>0 ? rsqrt : 0
__global__ void deg_rsqrt_kernel(const float* __restrict__ M, int n,
                                 float* __restrict__ d) {
    __shared__ float red[256];
    int row = blockIdx.x;
    float s = 0.0f;
    for (int j = threadIdx.x; j < n; j += 256) {
        float v = M[(size_t)row * n + j];
        if (j == row) v = (v == 0.0f) ? 2.0f : v;
        s += v;
    }
    red[threadIdx.x] = s;
    __syncthreads();
    for (int w = 128; w > 0; w >>= 1) {
        if (threadIdx.x < w) red[threadIdx.x] += red[threadIdx.x + w];
        __syncthreads();
    }
    if (threadIdx.x == 0) {
        float dg = red[0];
        d[row] = dg > 0.0f ? rsqrtf(dg) : 0.0f;
    }
}

// Ahat[i,j] = d[i]*d[j]*A[i,j]  (A = M with diag fixed to 2 where zero)
__global__ void build_ahat_kernel(const float* __restrict__ M,
                                  const float* __restrict__ d, int n,
                                  float* __restrict__ Ah) {
    size_t idx = (size_t)blockIdx.x * 256 + threadIdx.x;
    size_t total = (size_t)n * n;
    if (idx >= total) return;
    int i = (int)(idx / n), j = (int)(idx % n);
    float v = M[idx];
    if (i == j) v = (v == 0.0f) ? 2.0f : v;
    Ah[idx] = d[i] * d[j] * v;
}

// Md = M with diagonal forced to 1 (remove self loops, add eye)
__global__ void build_md_kernel(const float* __restrict__ M, int n,
                                float* __restrict__ Md) {
    size_t idx = (size_t)blockIdx.x * 256 + threadIdx.x;
    size_t total = (size_t)n * n;
    if (idx >= total) return;
    int i = (int)(idx / n), j = (int)(idx % n);
    Md[idx] = (i == j) ? 1.0f : M[idx];
}

// s[i] = tanh((x_i . p) / ||p||)
__global__ void score_kernel(const float* __restrict__ X,
                             const float* __restrict__ p, int n, int ch,
                             float* __restrict__ s) {
    int i = blockIdx.x * 256 + threadIdx.x;
    if (i >= n) return;
    float dot = 0.0f, nn = 0.0f;
    for (int c = 0; c < ch; ++c) {
        float pc = p[c];
        dot += X[(size_t)i * ch + c] * pc;
        nn += pc * pc;
    }
    s[i] = tanhf(dot * rsqrtf(nn));
}

// Deterministic top-k via exact rank (descending score, index tiebreak)
__global__ void rank_select_kernel(const float* __restrict__ s, int n, int k,
                                   int* __restrict__ perm) {
    int i = blockIdx.x * 256 + threadIdx.x;
    if (i >= n) return;
    float si = s[i];
    int r = 0;
    for (int j = 0; j < n; ++j) {
        float sj = s[j];
        r += (sj > si) || (sj == si && j < i);
    }
    if (r < k) perm[r] = i;
}

// Xo[r,c] = X[perm[r],c] * s[perm[r]]
__global__ void gather_pool_kernel(const float* __restrict__ X,
                                   const float* __restrict__ s,
                                   const int* __restrict__ perm,
                                   int k, int ch, float* __restrict__ Xo) {
    int idx = blockIdx.x * 256 + threadIdx.x;
    if (idx >= k * ch) return;
    int r = idx / ch, c = idx % ch;
    int src = perm[r];
    Xo[idx] = X[(size_t)src * ch + c] * s[src];
}

// Mo[r1,r2] = Mi[perm[r1], perm[r2]]
__global__ void gather_adj_kernel(const float* __restrict__ Mi,
                                  const int* __restrict__ perm,
                                  int nin, int k, float* __restrict__ Mo) {
    size_t idx = (size_t)blockIdx.x * 256 + threadIdx.x;
    size_t total = (size_t)k * k;
    if (idx >= total) return;
    int r = (int)(idx / k), c = (int)(idx % k);
    Mo[idx] = Mi[(size_t)perm[r] * nin + perm[c]];
}

// out[n, 2ch]: [:,0:ch] = res, [:,ch:2ch] = 0
__global__ void concat_res_kernel(const float* __restrict__ res, int n, int ch,
                                  float* __restrict__ out) {
    size_t idx = (size_t)blockIdx.x * 256 + threadIdx.x;
    size_t total = (size_t)n * 2 * ch;
    if (idx >= total) return;
    int i = (int)(idx / (2 * ch)), c = (int)(idx % (2 * ch));
    out[idx] = (c < ch) ? res[(size_t)i * ch + c] : 0.0f;
}

// out[perm[r], ch + c] = xup[r, c]
__global__ void scatter_up_kernel(const float* __restrict__ xup,
                                  const int* __restrict__ perm,
                                  int k, int ch, float* __restrict__ out) {
    int idx = blockIdx.x * 256 + threadIdx.x;
    if (idx >= k * ch) return;
    int r = idx / ch, c = idx % ch;
    out[(size_t)perm[r] * (2 * ch) + ch + c] = xup[idx];
}

// ---------------------------------------------------------------------------
// Host-side orchestration
// ---------------------------------------------------------------------------

static inline unsigned cdivu(size_t a, size_t b) { return (unsigned)((a + b - 1) / b); }

static void gemm(hipStream_t st, const float* A, const float* B, float* C,
                 int M, int N, int K, const float* bias, int relu, int zdiag) {
    dim3 grid(cdivu(M, GEMM_BM), cdivu(N, GEMM_BN));
    gemm_f32_wmma<<<grid, 256, 0, st>>>(A, B, C, M, N, K, bias, relu, zdiag);
}

// gcn: Xout = relu?( Ahat @ (Xin @ W) + b ), Ahat materialized into T1
static void gcn(hipStream_t st, const float* Madj, int n,
                const float* Xin, int chin, const float* W, int chout,
                const float* b, int relu,
                float* dv, float* T1, float* Y, float* Xout) {
    deg_rsqrt_kernel<<<n, 256, 0, st>>>(Madj, n, dv);
    size_t tot = (size_t)n * n;
    build_ahat_kernel<<<cdivu(tot, 256), 256, 0, st>>>(Madj, dv, n, T1);
    gemm(st, Xin, W, Y, n, chout, chin, nullptr, 0, 0);
    gemm(st, T1, Y, Xout, n, chout, n, b, relu, 0);
}

// augment: T2 = zero_diag( (M diag->1) @ (M diag->1) )
static void augment(hipStream_t st, const float* Madj, int n, float* T1, float* T2) {
    size_t tot = (size_t)n * n;
    build_md_kernel<<<cdivu(tot, 256), 256, 0, st>>>(Madj, n, T1);
    gemm(st, T1, T1, T2, n, n, n, nullptr, 0, /*zero_diag=*/1);
}

static void pool(hipStream_t st, const float* Xin, int n, int ch, const float* p,
                 const float* Maug, int k, float* sv, int* perm,
                 float* Xout, float* Mout) {
    score_kernel<<<cdivu(n, 256), 256, 0, st>>>(Xin, p, n, ch, sv);
    rank_select_kernel<<<cdivu(n, 256), 256, 0, st>>>(sv, n, k, perm);
    gather_pool_kernel<<<cdivu((size_t)k * ch, 256), 256, 0, st>>>(Xin, sv, perm, k, ch, Xout);
    size_t tot = (size_t)k * k;
    gather_adj_kernel<<<cdivu(tot, 256), 256, 0, st>>>(Maug, perm, n, k, Mout);
}

extern "C" void kernel_launch(void* const* d_in, const int* in_sizes, int n_in,
                              void* d_out, int out_size, void* d_ws, size_t ws_size,
                              hipStream_t stream) {
    (void)n_in; (void)out_size; (void)ws_size;

    constexpr int N0 = 3072, N1 = 1536, N2 = 768, N3 = 384;
    constexpr int CH0 = 128, CH1 = 256, CH2 = 256, OUTC = 128, NCLS = 8, INC = 16;

    // Inputs (setup_inputs dict order)
    const float* x_in  = (const float*)d_in[0];
    const int*   ei    = (const int*)d_in[1];
    const int    E     = in_sizes[1] / 2;
    const float* Wd0 = (const float*)d_in[2];  const float* bd0 = (const float*)d_in[3];
    const float* Wd1 = (const float*)d_in[4];  const float* bd1 = (const float*)d_in[5];
    const float* Wd2 = (const float*)d_in[6];  const float* bd2 = (const float*)d_in[7];
    const float* Wd3 = (const float*)d_in[8];  const float* bd3 = (const float*)d_in[9];
    const float* p0  = (const float*)d_in[10];
    const float* p1  = (const float*)d_in[11];
    const float* p2  = (const float*)d_in[12];
    const float* Wu0 = (const float*)d_in[13]; const float* bu0 = (const float*)d_in[14];
    const float* Wu1 = (const float*)d_in[15]; const float* bu1 = (const float*)d_in[16];
    const float* Wu2 = (const float*)d_in[17]; const float* bu2 = (const float*)d_in[18];
    const float* Wdec = (const float*)d_in[19]; const float* bdec = (const float*)d_in[20];
    float* out = (float*)d_out;

    // Workspace bump allocation (floats)
    float* ws = (float*)d_ws;
    size_t off = 0;
    float* M0  = ws + off; off += (size_t)N0 * N0;   // Ms[0], preserved
    float* M1  = ws + off; off += (size_t)N1 * N1;   // Ms[1]
    float* M2s = ws + off; off += (size_t)N2 * N2;   // Ms[2]
    float* M3  = ws + off; off += (size_t)N3 * N3;   // deepest M
    float* T1  = ws + off; off += (size_t)N0 * N0;   // Md / Ahat scratch
    float* T2  = ws + off; off += (size_t)N0 * N0;   // augment result
    float* X0  = ws + off; off += (size_t)N0 * CH0;  // xs[0]
    float* X1  = ws + off; off += (size_t)N1 * CH1;  // xs[1]
    float* X2  = ws + off; off += (size_t)N2 * CH2;  // xs[2]
    float* XA  = ws + off; off += (size_t)N0 * 512;  // ping
    float* XB  = ws + off; off += (size_t)N0 * 512;  // pong
    float* dv  = ws + off; off += N0;                // rsqrt(deg)
    float* sv  = ws + off; off += N0;                // scores
    int* perm0 = (int*)(ws + off); off += N1;
    int* perm1 = (int*)(ws + off); off += N2;
    int* perm2 = (int*)(ws + off); off += N3;

    // Build dense adjacency M0[dst, src] += 1
    size_t m0tot = (size_t)N0 * N0;
    fill_zero_kernel<<<cdivu(m0tot, 256), 256, 0, stream>>>(M0, m0tot);
    scatter_edges_kernel<<<cdivu(E, 256), 256, 0, stream>>>(ei, E, N0, M0);

    // Down path
    gcn(stream, M0, N0, x_in, INC, Wd0, CH0, bd0, /*relu=*/1, dv, T1, XB, X0);

    augment(stream, M0, N0, T1, T2);
    pool(stream, X0, N0, CH0, p0, T2, N1, sv, perm0, XA, M1);
    gcn(stream, M1, N1, XA, CH0, Wd1, CH1, bd1, 1, dv, T1, XB, X1);

    augment(stream, M1, N1, T1, T2);
    pool(stream, X1, N1, CH1, p1, T2, N2, sv, perm1, XA, M2s);
    gcn(stream, M2s, N2, XA, CH1, Wd2, CH2, bd2, 1, dv, T1, XB, X2);

    augment(stream, M2s, N2, T1, T2);
    pool(stream, X2, N2, CH2, p2, T2, N3, sv, perm2, XA, M3);
    gcn(stream, M3, N3, XA, CH2, Wd3, CH2, bd3, 1, dv, T1, XB, XA);
    // x = XA (384 x 256)

    // Up path, i=0 (j=2): concat [X2 | unpool(XA)] -> XB (768 x 512)
    concat_res_kernel<<<cdivu((size_t)N2 * 2 * CH2, 256), 256, 0, stream>>>(X2, N2, CH2, XB);
    scatter_up_kernel<<<cdivu((size_t)N3 * CH2, 256), 256, 0, stream>>>(XA, perm2, N3, CH2, XB);
    gcn(stream, M2s, N2, XB, 2 * CH2, Wu0, CH1, bu0, 1, dv, T1, XA, XB);
    // x = XB (768 x 256)

    // i=1 (j=1): concat [X1 | unpool(XB)] -> XA (1536 x 512)
    concat_res_kernel<<<cdivu((size_t)N1 * 2 * CH1, 256), 256, 0, stream>>>(X1, N1, CH1, XA);
    scatter_up_kernel<<<cdivu((size_t)N2 * CH1, 256), 256, 0, stream>>>(XB, perm1, N2, CH1, XA);
    gcn(stream, M1, N1, XA, 2 * CH1, Wu1, OUTC, bu1, 1, dv, T1, XB, XA);
    // x = XA (1536 x 128)

    // i=2 (j=0): concat [X0 | unpool(XA)] -> XB (3072 x 256)
    concat_res_kernel<<<cdivu((size_t)N0 * 2 * CH0, 256), 256, 0, stream>>>(X0, N0, CH0, XB);
    scatter_up_kernel<<<cdivu((size_t)N1 * CH0, 256), 256, 0, stream>>>(XA, perm0, N1, CH0, XB);
    gcn(stream, M0, N0, XB, 2 * CH0, Wu2, OUTC, bu2, /*relu=*/0, dv, T1, XA, XB);
    // x = XB (3072 x 128)

    // Decode: out = x @ Wdec + bdec  (3072 x 8)
    gemm(stream, XB, Wdec, out, N0, NCLS, OUTC, bdec, 0, 0);
}